// SupPixPool_5669356835625
// MI455X (gfx1250) — compile-verified
//
#include <hip/hip_runtime.h>
#include <hip/hip_bf16.h>
#include <math.h>

// Problem constants (match reference setup_inputs)
constexpr int B      = 4;
constexpr int C      = 64;
constexpr int H      = 512;
constexpr int W      = 512;
constexpr int HW     = H * W;      // 262144
constexpr int K      = 1024;       // number of segments

// Tiling
constexpr int CG     = 8;          // channels per block (LDS table = CG*K*4 = 32 KB)
constexpr int NCG    = C / CG;     // 8 channel groups
constexpr int TILES  = 32;         // HW tiles per (b, channel-group)
constexpr int TILE   = HW / TILES; // 8192 pixels per block
constexpr int NT     = 256;        // threads per block (8 wave32 waves)

// Monotone bijection: IEEE-754 float total order  <->  unsigned integer order.
//   bits >= 0 (sign clear): u =  bits ^ 0x80000000   (positives -> [0x8000_0000, ...])
//   bits <  0 (sign set)  : u = ~bits               (negatives -> [0, 0x7FFF_FFFF])
// max over u == max over float. Branchless: mask = (bits>>31) | 0x80000000.
__device__ __forceinline__ unsigned int enc_f32(float v) {
    const int bits = __float_as_int(v);
    return (unsigned int)(bits ^ ((bits >> 31) | 0x80000000));
}
__device__ __forceinline__ float dec_f32(unsigned int u) {
    const int m = ((int)u >> 31);                    // -1 if MSB set, 0 otherwise
    return __int_as_float((int)(u ^ (unsigned int)(~m | 0x80000000)));
}

// encode(-inf): -inf bits = 0xFF800000 -> u = ~bits = 0x007FFFFF.
constexpr unsigned int ENC_NEG_INF = 0x007FFFFFu;

__global__ void __launch_bounds__(NT)
spxpool_init_kernel(unsigned int* __restrict__ out_u, int n) {
    const int i = blockIdx.x * NT + threadIdx.x;
    if (i < n) out_u[i] = ENC_NEG_INF;
}

__global__ void __launch_bounds__(NT)
spxpool_max_kernel(const float* __restrict__ img,
                   const int*   __restrict__ spx,
                   unsigned int* __restrict__ out_u) {
    __shared__ unsigned int smax[CG * K];   // 32 KB private segment-max table (encoded)

    const int tid = threadIdx.x;

    // Init private table to the (encoded) max identity
    #pragma unroll
    for (int i = tid; i < CG * K; i += NT) smax[i] = ENC_NEG_INF;
    __syncthreads();

    const int tile = blockIdx.x;     // 0..TILES-1
    const int cg   = blockIdx.y;     // 0..NCG-1
    const int b    = blockIdx.z;     // 0..B-1
    const int hw0  = tile * TILE;

    const float* imgB = img + (size_t)(b * C + cg * CG) * HW + hw0;
    const int*   spxB = spx + (size_t)b * HW + hw0;

    // Stream the tile: each thread handles 4 consecutive pixels per step
    // (b128 loads), 8 steps total. LDS layout: smax[j*K + k].
    for (int it = tid * 4; it < TILE; it += NT * 4) {
        const int4 s4 = *reinterpret_cast<const int4*>(spxB + it);

        // One light speculative prefetch of the next label line
        // (global_prefetch_b8 on gfx1250).
        const int itn = it + NT * 4;
        if (itn < TILE) __builtin_prefetch(spxB + itn, 0, 0);

        #pragma unroll
        for (int j = 0; j < CG; ++j) {
            const float4 v = *reinterpret_cast<const float4*>(imgB + (size_t)j * HW + it);
            atomicMax(&smax[j * K + s4.x], enc_f32(v.x));   // single ds_max_u32 each
            atomicMax(&smax[j * K + s4.y], enc_f32(v.y));
            atomicMax(&smax[j * K + s4.z], enc_f32(v.z));
            atomicMax(&smax[j * K + s4.w], enc_f32(v.w));
        }
    }
    __syncthreads();

    // Merge private table into global output (encoded space).
    // smax index i == (local channel j)*K + k maps directly onto outB + i.
    unsigned int* outB = out_u + (size_t)(b * C + cg * CG) * K;
    #pragma unroll
    for (int i = tid; i < CG * K; i += NT) {
        const unsigned int u = smax[i];
        if (u != ENC_NEG_INF) {
            atomicMax(outB + i, u);                          // global_atomic_max_u32
        }
    }
}

__global__ void __launch_bounds__(NT)
spxpool_finalize_kernel(unsigned int* __restrict__ buf, int n) {
    const int i = blockIdx.x * NT + threadIdx.x;
    if (i < n) {
        const unsigned int u = buf[i];
        reinterpret_cast<float*>(buf)[i] = dec_f32(u);       // in-place decode to float
    }
}

extern "C" void kernel_launch(void* const* d_in, const int* in_sizes, int n_in,
                              void* d_out, int out_size, void* d_ws, size_t ws_size,
                              hipStream_t stream) {
    (void)in_sizes; (void)n_in; (void)d_ws; (void)ws_size;

    const float*  img   = reinterpret_cast<const float*>(d_in[0]);
    const int*    spx   = reinterpret_cast<const int*>(d_in[1]);
    unsigned int* out_u = reinterpret_cast<unsigned int*>(d_out);

    const int n      = out_size;               // B*C*K = 262144
    const int blocks = (n + NT - 1) / NT;

    // 1) out = encode(-inf) (segment_max identity; harness poisons d_out)
    spxpool_init_kernel<<<blocks, NT, 0, stream>>>(out_u, n);

    // 2) LDS-privatized segment max (encoded u32), global merge via atomic max
    {
        dim3 grid(TILES, NCG, B);              // 32 x 8 x 4 = 1024 blocks
        spxpool_max_kernel<<<grid, NT, 0, stream>>>(img, spx, out_u);
    }

    // 3) Decode encoded maxima back to IEEE float in place
    spxpool_finalize_kernel<<<blocks, NT, 0, stream>>>(out_u, n);
}